// ConvexSampler_72481868087352
// MI455X (gfx1250) — compile-verified
//
#include <hip/hip_runtime.h>
#include <hip/hip_bf16.h>
#include <stdint.h>

#define AS1 __attribute__((address_space(1)))
#define AS3 __attribute__((address_space(3)))

typedef int v4i __attribute__((ext_vector_type(4)));

#define CPY_THREADS 256
#define CPY_BATCH   4
#define CPY_F4_PER_BLOCK (CPY_THREADS * CPY_BATCH)   // 1024 float4 = 16 KiB per phase

// ---- gfx1250 async global<->LDS builtins (existence confirmed by probes) ----
#if defined(__HIP_DEVICE_COMPILE__)
#if __has_builtin(__builtin_amdgcn_global_load_async_to_lds_b128) && \
    __has_builtin(__builtin_amdgcn_global_store_async_from_lds_b128)
#define ASYNC_BUILTINS 1
#else
#define ASYNC_BUILTINS 0
#endif
#else
#define ASYNC_BUILTINS 0
#endif

__device__ __forceinline__ void wait_async_0() {
#if defined(__HIP_DEVICE_COMPILE__)
#if __has_builtin(__builtin_amdgcn_s_wait_asynccnt)
  __builtin_amdgcn_s_wait_asynccnt(0);
#else
  asm volatile("s_wait_asynccnt 0" ::: "memory");
#endif
#endif
}

// one 16B lane-transfer: global -> LDS (ASYNCcnt path)
__device__ __forceinline__ void async_ld_g2l(const float4* g, unsigned lds_off) {
#if ASYNC_BUILTINS
  __builtin_amdgcn_global_load_async_to_lds_b128(
      (AS1 v4i*)(uintptr_t)g, (AS3 v4i*)(uintptr_t)lds_off, 0, 0);
#else
  unsigned long long ga = (unsigned long long)(uintptr_t)g;
  asm volatile("global_load_async_to_lds_b128 %0, %1, off"
               :: "v"(lds_off), "v"(ga) : "memory");
#endif
}

// one 16B lane-transfer: LDS -> global (ASYNCcnt path)
__device__ __forceinline__ void async_st_l2g(float4* g, unsigned lds_off) {
#if ASYNC_BUILTINS
  __builtin_amdgcn_global_store_async_from_lds_b128(
      (AS1 v4i*)(uintptr_t)g, (AS3 v4i*)(uintptr_t)lds_off, 0, 0);
#else
  unsigned long long ga = (unsigned long long)(uintptr_t)g;
  asm volatile("global_store_async_from_lds_b128 %0, %1, off"
               :: "v"(ga), "v"(lds_off) : "memory");
#endif
}

// --------------------------------------------------------------------------
// Kernel 1: bulk copy z -> out[0 : B*D) via async LDS staging (B128, wave32)
// --------------------------------------------------------------------------
__global__ void __launch_bounds__(CPY_THREADS)
bulk_copy_kernel(const float4* __restrict__ src, float4* __restrict__ dst,
                 long long n4) {
#if defined(__gfx1250__)
  __shared__ alignas(16) float4 stage[CPY_F4_PER_BLOCK];
  const unsigned tid      = threadIdx.x;
  const unsigned lds_base = (unsigned)(uintptr_t)&stage[0];  // low 32b of generic = LDS offset
  long long       base    = (long long)blockIdx.x * CPY_F4_PER_BLOCK;
  const long long stride  = (long long)gridDim.x * CPY_F4_PER_BLOCK;
  for (; base < n4; base += stride) {
#pragma unroll
    for (int j = 0; j < CPY_BATCH; ++j) {
      const long long idx = base + (long long)j * CPY_THREADS + tid;
      if (idx < n4)
        async_ld_g2l(src + idx, lds_base + (unsigned)(j * CPY_THREADS + tid) * 16u);
    }
    wait_async_0();  // loads have landed in our private LDS slots
#pragma unroll
    for (int j = 0; j < CPY_BATCH; ++j) {
      const long long idx = base + (long long)j * CPY_THREADS + tid;
      if (idx < n4)
        async_st_l2g(dst + idx, lds_base + (unsigned)(j * CPY_THREADS + tid) * 16u);
    }
    wait_async_0();  // stores done before LDS slots are reused
  }
#else
  // host pass / non-gfx1250: plain vector copy so the file always parses+compiles
  long long i      = (long long)blockIdx.x * blockDim.x + threadIdx.x;
  long long stride = (long long)gridDim.x * blockDim.x;
  for (; i < n4; i += stride) dst[i] = src[i];
#endif
}

// --------------------------------------------------------------------------
// Kernel 2: 50 convex-combination rows; one block per row (L2-resident reads)
// --------------------------------------------------------------------------
__global__ void convex_kernel(const float* __restrict__ z, const int* __restrict__ pair,
                              const float* __restrict__ s, float* __restrict__ out,
                              int B, int D) {
  const int   row = blockIdx.x;
  const int   ia  = pair[2 * row + 0];
  const int   ib  = pair[2 * row + 1];
  const float sv  = s[row];
  const float tv  = 1.0f - sv;
  const float* za = z + (long long)ia * D;
  const float* zb = z + (long long)ib * D;
  float*       o  = out + (long long)(B + row) * D;
  if ((D & 3) == 0) {
    const float4* a4 = (const float4*)za;
    const float4* b4 = (const float4*)zb;
    float4*       o4 = (float4*)o;
    const int     d4 = D >> 2;
    for (int j = threadIdx.x; j < d4; j += blockDim.x) {
      const float4 a = a4[j], b = b4[j];
      float4 r;
      r.x = sv * a.x + tv * b.x;
      r.y = sv * a.y + tv * b.y;
      r.z = sv * a.z + tv * b.z;
      r.w = sv * a.w + tv * b.w;
      o4[j] = r;
    }
  } else {
    for (int j = threadIdx.x; j < D; j += blockDim.x)
      o[j] = sv * za[j] + tv * zb[j];
  }
}

// --------------------------------------------------------------------------
// Kernel 3: labels_out (cast int->float, OOS fill) + data_type_out (+ tail)
// --------------------------------------------------------------------------
__global__ void tails_kernel(const int* __restrict__ labels, const float* __restrict__ dtype,
                             const int* __restrict__ oosp, float* __restrict__ out,
                             int B, int N, long long zcount,
                             const float* __restrict__ z, long long BD) {
  const long long T = (long long)B + N;
  const long long i = (long long)blockIdx.x * blockDim.x + threadIdx.x;
  if (i == 0) {  // scalar remainder of the bulk copy (BD % 4), zero for these shapes
    for (long long r = (BD >> 2) << 2; r < BD; ++r) out[r] = z[r];
  }
  if (i >= T) return;
  const float lab = (i < B) ? (float)labels[i] : (float)(*oosp);
  const float dt  = (i < B) ? dtype[i]         : 1.0f;
  out[zcount + i]     = lab;  // labels_out
  out[zcount + T + i] = dt;   // data_type_out
}

// --------------------------------------------------------------------------
extern "C" void kernel_launch(void* const* d_in, const int* in_sizes, int n_in,
                              void* d_out, int out_size, void* d_ws, size_t ws_size,
                              hipStream_t stream) {
  const float* z      = (const float*)d_in[0];
  const int*   labels = (const int*)  d_in[1];
  const float* dtype  = (const float*)d_in[2];
  const int*   pair   = (const int*)  d_in[3];
  const float* s      = (const float*)d_in[4];
  const int*   oosp   = (const int*)  d_in[5];
  float*       out    = (float*)d_out;

  const long long BD = in_sizes[0];          // 16384*2048
  const int       B  = in_sizes[1];          // 16384
  const int       N  = in_sizes[4];          // 50
  const int       D  = (int)(BD / B);        // 2048
  const long long n4 = BD >> 2;              // float4 count
  const long long zc = (long long)(B + N) * D;

  long long gridA = (n4 + CPY_F4_PER_BLOCK - 1) / CPY_F4_PER_BLOCK;
  if (gridA > 8192) gridA = 8192;
  bulk_copy_kernel<<<(int)gridA, CPY_THREADS, 0, stream>>>(
      (const float4*)z, (float4*)out, n4);

  if (N > 0)
    convex_kernel<<<N, 256, 0, stream>>>(z, pair, s, out, B, D);

  const int T = B + N;
  tails_kernel<<<(T + 255) / 256, 256, 0, stream>>>(
      labels, dtype, oosp, out, B, N, zc, z, BD);
}